// CustomLoss_14242111553840
// MI455X (gfx1250) — compile-verified
//
#include <hip/hip_runtime.h>
#include <math.h>

// ---------------------------------------------------------------------------
// CustomLoss on MI455X (gfx1250, wave32).
// Bandwidth-bound fp32: 262 MB read once -> ~11.3 us floor @ 23.3 TB/s.
// One wave per row (1000 fp32 = exactly 250 float4 -> 8 b128 loads/lane),
// row cached in registers, two logical passes, no re-read from memory.
// All logs/exps kept in base-2 domain on raw v_log_f32 / v_exp_f32 (inputs
// provably never denormal), scaled by ln2 once at the very end.
// CE term uses a uniform scalar re-load of x_t (L2 hit) instead of a
// per-element select chain; accumulated by all 32 lanes, /32 at the end.
// Cross-lane sums use V_WMMA_F32_16X16X4_F32 (A = lane partials, B = ones).
// Deterministic two-stage reduction (no float atomics).
// ---------------------------------------------------------------------------

typedef __attribute__((ext_vector_type(2))) float v2f;
typedef __attribute__((ext_vector_type(8))) float v8f;

#define NUM_CLASSES     1000
#define VEC4            (NUM_CLASSES / 4)   // 250, exact
#define LAMBDA_F        0.5f
#define ROWS_PER_WAVE   8
#define WAVES_PER_BLOCK 8
#define THREADS         (WAVES_PER_BLOCK * 32)

#define LOG2E_F 1.44269504088896340736f
#define LN2_F   0.69314718055994530942f

// Wave32 all-lanes sum via WMMA f32 16x16x4:
//   A (16x4 f32): vgpr0 = v (K=0 lanes0-15 / K=2 lanes16-31), vgpr1 = 0
//   B (4x16 f32): all ones
//   D[m,n] = v[m] + v[m+16]  (independent of n)
//   per-lane sum of 8 D vgprs = half sum; shfl_xor(16) completes it.
// Requires EXEC == all 1s (all call sites are at full, reconverged EXEC).
__device__ __forceinline__ float wave_sum_f32_wmma(float v) {
  v2f a; a[0] = v;    a[1] = 0.0f;
  v2f b; b[0] = 1.0f; b[1] = 1.0f;
  v8f c = {};
  v8f d = __builtin_amdgcn_wmma_f32_16x16x4_f32(
      /*neg_a=*/false, a, /*neg_b=*/false, b,
      /*c_mod=*/(short)0, c, /*reuse_a=*/false, /*reuse_b=*/false);
  float s = (d[0] + d[1]) + (d[2] + d[3]);
  s += (d[4] + d[5]) + (d[6] + d[7]);
  s += __shfl_xor(s, 16, 32);
  return s;
}

__device__ __forceinline__ float wave_max_f32(float v) {
#pragma unroll
  for (int off = 16; off > 0; off >>= 1)
    v = fmaxf(v, __shfl_xor(v, off, 32));
  return v;
}

__global__ __launch_bounds__(THREADS) void loss_partial_kernel(
    const float* __restrict__ outputs, const int* __restrict__ targets,
    float* __restrict__ partials, int B) {
  const int lane = threadIdx.x & 31;
  const int wid  = threadIdx.x >> 5;
  const int gwave = blockIdx.x * WAVES_PER_BLOCK + wid;

  float lp = 0.0f;  // 32x sum of log2 p_t        (* ln2/32 at the end)
  float un = 0.0f;  // sum of (t-j)*log2(1-p_j)   (* ln2/C at the end)
  float ov = 0.0f;  // sum of log2(1-p_j), j>t    (* ln2 at the end)

  const int row0 = gwave * ROWS_PER_WAVE;
  for (int r = 0; r < ROWS_PER_WAVE; ++r) {
    const int row = row0 + r;
    if (row >= B) break;  // uniform across the wave

    const float* rowp = outputs + (size_t)row * NUM_CLASSES;
    const float4* rowv = reinterpret_cast<const float4*>(rowp);
    const int   t  = targets[row];        // wave-uniform -> scalar load
    const float tf = (float)t;
    const float xt = rowp[t];             // uniform target logit (L2 hit)

    // ---- load entire row into registers (coalesced b128) ----
    float4 xv[8];
#pragma unroll
    for (int k = 0; k < 8; ++k) {
      const int v = lane + 32 * k;          // k<7 provably in range
      if (v < VEC4) xv[k] = rowv[v];
      else          xv[k] = make_float4(-1e30f, -1e30f, -1e30f, -1e30f);
    }

    // ---- pass 1a: row max ----
    float m = -1e30f;
#pragma unroll
    for (int k = 0; k < 8; ++k)
      m = fmaxf(m, fmaxf(fmaxf(xv[k].x, xv[k].y), fmaxf(xv[k].z, xv[k].w)));
    m = wave_max_f32(m);

    // ---- pass 1b: e = 2^(x*log2e - m*log2e) in place + sumexp ----
    // (one v_fma + one v_exp per element; -1e30 padding underflows to 0)
    const float m2 = m * LOG2E_F;
    float esum = 0.0f;
#pragma unroll
    for (int k = 0; k < 8; ++k) {
      xv[k].x = __builtin_amdgcn_exp2f(__builtin_fmaf(xv[k].x, LOG2E_F, -m2));
      xv[k].y = __builtin_amdgcn_exp2f(__builtin_fmaf(xv[k].y, LOG2E_F, -m2));
      xv[k].z = __builtin_amdgcn_exp2f(__builtin_fmaf(xv[k].z, LOG2E_F, -m2));
      xv[k].w = __builtin_amdgcn_exp2f(__builtin_fmaf(xv[k].w, LOG2E_F, -m2));
      esum += (xv[k].x + xv[k].y) + (xv[k].z + xv[k].w);
    }
    const float s     = wave_sum_f32_wmma(esum);   // WMMA reduce, broadcast
    const float rcp_s = __builtin_amdgcn_rcpf(s);
    const float ls2   = __builtin_amdgcn_logf(s);  // log2(s), once per row

    // CE (uniform, all 32 lanes accumulate; /32 folded into final scale):
    // log2(p_t) = (x_t - m)*log2e - log2(s)
    lp += __builtin_fmaf(xt, LOG2E_F, -m2) - ls2;

    // ---- pass 2: branchless accumulate.
    // Padding lanes: e=0 -> 1-p=1 -> l2=0; j>=1000>t -> wd<0 -> max()=0. ----
#pragma unroll
    for (int k = 0; k < 8; ++k) {
      const float jf0 = (float)(4 * (lane + 32 * k));
#define PROC_ELEM(E_, CC_)                                                 \
      {                                                                    \
        const float wd = tf - (jf0 + (float)(CC_));  /* t - j, exact */    \
        const float l2 = __builtin_amdgcn_logf(                            \
            __builtin_fmaf(-(E_), rcp_s, 1.0f));     /* log2(1-p) */       \
        un  = __builtin_fmaf(fmaxf(wd, 0.0f), l2, un);                     \
        ov += (wd < 0.0f) ? l2 : 0.0f;                                     \
      }
      PROC_ELEM(xv[k].x, 0)
      PROC_ELEM(xv[k].y, 1)
      PROC_ELEM(xv[k].z, 2)
      PROC_ELEM(xv[k].w, 3)
#undef PROC_ELEM
    }
  }

  // ---- wave reduction (WMMA) + block reduction (LDS) ----
  lp = wave_sum_f32_wmma(lp);
  un = wave_sum_f32_wmma(un);
  ov = wave_sum_f32_wmma(ov);

  __shared__ float sred[WAVES_PER_BLOCK][3];
  if (lane == 0) { sred[wid][0] = lp; sred[wid][1] = un; sred[wid][2] = ov; }
  __syncthreads();
  if (threadIdx.x < 3) {
    float acc = 0.0f;
#pragma unroll
    for (int w = 0; w < WAVES_PER_BLOCK; ++w) acc += sred[w][threadIdx.x];
    partials[blockIdx.x * 3 + threadIdx.x] = acc;
  }
}

__global__ __launch_bounds__(256) void loss_finalize_kernel(
    const float* __restrict__ partials, float* __restrict__ out,
    int nblk, float invB) {
  float lp = 0.0f, un = 0.0f, ov = 0.0f;
  for (int i = threadIdx.x; i < nblk; i += 256) {
    lp += partials[3 * i + 0];
    un += partials[3 * i + 1];
    ov += partials[3 * i + 2];
  }
  // loop fully reconverges -> EXEC all 1s at the WMMA reducers
  lp = wave_sum_f32_wmma(lp);
  un = wave_sum_f32_wmma(un);
  ov = wave_sum_f32_wmma(ov);

  __shared__ float sred[8][3];
  const int lane = threadIdx.x & 31;
  const int wid  = threadIdx.x >> 5;
  if (lane == 0) { sred[wid][0] = lp; sred[wid][1] = un; sred[wid][2] = ov; }
  __syncthreads();
  if (threadIdx.x == 0) {
    float LP = 0.0f, UN = 0.0f, OV = 0.0f;
#pragma unroll
    for (int w = 0; w < 8; ++w) {
      LP += sred[w][0]; UN += sred[w][1]; OV += sred[w][2];
    }
    // log2-domain -> ln; LP was accumulated by all 32 lanes -> /32 (exact)
    const float ce    = -LP * (LN2_F / 32.0f)                   * invB;
    const float under =  UN * (LN2_F / (float)NUM_CLASSES)      * invB;
    const float over  =  OV * (LN2_F)                           * invB;
    out[0] = ce - LAMBDA_F * (over + under);
  }
}

extern "C" void kernel_launch(void* const* d_in, const int* in_sizes, int n_in,
                              void* d_out, int out_size, void* d_ws, size_t ws_size,
                              hipStream_t stream) {
  const float* outputs = (const float*)d_in[0];
  const int*   targets = (const int*)d_in[1];
  const int B = in_sizes[1];  // 65536 rows

  const int waves = (B + ROWS_PER_WAVE - 1) / ROWS_PER_WAVE;
  const int nblk  = (waves + WAVES_PER_BLOCK - 1) / WAVES_PER_BLOCK;  // 1024

  float* partials = (float*)d_ws;  // 3 * nblk floats, fully rewritten each call

  loss_partial_kernel<<<nblk, THREADS, 0, stream>>>(outputs, targets, partials, B);
  loss_finalize_kernel<<<1, 256, 0, stream>>>(partials, (float*)d_out, nblk,
                                              1.0f / (float)B);
}